// ConvolutionalCapsule_55001351192880
// MI455X (gfx1250) — compile-verified
//
#include <hip/hip_runtime.h>
#include <math.h>

typedef __attribute__((ext_vector_type(2))) float v2f;
typedef __attribute__((ext_vector_type(8))) float v8f;
typedef __attribute__((ext_vector_type(4))) int   v4i;

#define EPS_F 1e-9f
#define BB   8
#define HH   14
#define WWD  14
#define OHW  12
#define II   144   // 3*3*16 input capsules
#define OO   16
#define PPC  16    // pose elements
#define VSTR 257   // padded votes row stride (floats); 257 mod 64 == 1 -> conflict-free

#if defined(__has_builtin)
#if __has_builtin(__builtin_amdgcn_global_load_async_to_lds_b128) && \
    __has_builtin(__builtin_amdgcn_s_wait_asynccnt)
#define USE_ASYNC_LDS 1
typedef __attribute__((address_space(1))) v4i glob_v4i;
typedef __attribute__((address_space(3))) v4i lds_v4i;
#endif
#endif

// One workgroup per output spatial position n = (b, oy, ox).
// Stage 1: 3x3 patch of pose (9 x 1KB rows) + activations into LDS
//          (async path: GLOBAL_LOAD_ASYNC_TO_LDS_B128, tracked by ASYNCcnt).
// Stage 2: votes[i,o,x,z] = sum_y pose[i,x,y]*w[i,o,y,z] via V_WMMA_F32_16X16X4_F32,
//          votes tile (148 KB) stays in LDS -- never round-trips HBM.
// Stage 3: 3-iteration EM routing entirely in LDS.
__global__ __launch_bounds__(256)
void capsule_em_fused(const float* __restrict__ pose_in,  // [8,14,14,256]
                      const float* __restrict__ act_in,   // [8,14,14,16]
                      const float* __restrict__ w,        // [144,16,4,4]
                      const float* __restrict__ beta_v,   // [16]
                      const float* __restrict__ beta_a,   // [16]
                      float* __restrict__ out)            // pose 1152*256, then act 1152*16
{
    __shared__ float sm_votes[II * VSTR];                  // 148032 B
    __shared__ __align__(16) float sm_pose_rr[II * PPC];   // 9216 B: pose, later rr[i][o]
    __shared__ float sm_a[II];
    __shared__ float sm_mean[256];
    __shared__ float sm_inv2v[256];    // 1/(2*var+eps)
    __shared__ float sm_logstd[256];   // log(sqrt(var)+eps)
    __shared__ float sm_rsum[16];
    __shared__ float sm_oact[16];
    __shared__ float sm_logact[16];    // log(oact+eps) - sum_p log(std+eps)

    const int tid = threadIdx.x;
    const int n   = blockIdx.x;
    const int b   = n / (OHW * OHW);
    const int rem = n % (OHW * OHW);
    const int oy  = rem / OHW;
    const int ox  = rem % OHW;

    // ---------------- Stage 1: patch -> LDS ----------------
#ifdef USE_ASYNC_LDS
    // 9 patch rows x 1KB contiguous -> 576 B128 async transfers straight to LDS.
    for (int idx = tid; idx < 9 * 64; idx += 256) {
        const int seg = idx >> 6;           // patch 0..8
        const int off = idx & 63;           // 16B unit within the 1KB row
        const int ky = seg / 3, kx = seg % 3;
        const float* g = pose_in
            + (size_t)(((b * HH + oy + ky) * WWD) + ox + kx) * 256 + off * 4;
        __builtin_amdgcn_global_load_async_to_lds_b128(
            (glob_v4i*)(g),
            (lds_v4i*)(&sm_pose_rr[idx * 4]),
            0, 0);
    }
#else
    for (int idx = tid; idx < II * PPC; idx += 256) {
        int i = idx >> 4, p = idx & 15;
        int patch = i >> 4, c = i & 15;          // i = (ky*3+kx)*16 + c
        int ky = patch / 3, kx = patch % 3;
        sm_pose_rr[idx] =
            pose_in[((b * HH + oy + ky) * WWD + ox + kx) * 256 + c * 16 + p];
    }
#endif
    for (int i = tid; i < II; i += 256) {
        int patch = i >> 4, c = i & 15;
        int ky = patch / 3, kx = patch % 3;
        sm_a[i] = act_in[((b * HH + oy + ky) * WWD + ox + kx) * 16 + c];
    }
#ifdef USE_ASYNC_LDS
    __builtin_amdgcn_s_wait_asynccnt(0);
#endif
    __syncthreads();

    // ---------------- Stage 2: votes via fp32 WMMA ----------------
    const int wave  = tid >> 5;
    const int lane  = tid & 31;
    const int mrow  = lane & 15;        // A row M / B,D column
    const int k0    = (lane >> 4) * 2;  // lanes 0-15: K=0,1 ; lanes 16-31: K=2,3
    const float amask = (mrow < 4) ? 1.0f : 0.0f;

    for (int i = wave; i < II; i += 8) {         // 18 uniform iterations per wave
        v2f A;
        A.x = sm_pose_rr[i * 16 + (mrow & 3) * 4 + k0]     * amask;
        A.y = sm_pose_rr[i * 16 + (mrow & 3) * 4 + k0 + 1] * amask;
        const int oL = mrow >> 2;   // o within o-group (B/D col = oL*4 + z)
        const int z  = mrow & 3;
        // prefetch all 4 o-groups' B operands as one load clause
        float b0[4], b1[4];
#pragma unroll
        for (int og = 0; og < 4; ++og) {
            const int o = og * 4 + oL;
            b0[og] = w[i * 256 + o * 16 + k0 * 4 + z];        // w[i,o,k0,z]
            b1[og] = w[i * 256 + o * 16 + (k0 + 1) * 4 + z];  // w[i,o,k0+1,z]
        }
#pragma unroll
        for (int og = 0; og < 4; ++og) {
            const int o = og * 4 + oL;
            v2f Bv; Bv.x = b0[og]; Bv.y = b1[og];
            v8f c0 = {};
            v8f d = __builtin_amdgcn_wmma_f32_16x16x4_f32(
                false, A, false, Bv, (short)0, c0, false, false);
            if (lane < 16) {                 // rows M=0..3 live in d[0..3], lanes 0-15
                const int base = i * VSTR + o * 16 + z;   // + x*4
                sm_votes[base + 0]  = d[0];
                sm_votes[base + 4]  = d[1];
                sm_votes[base + 8]  = d[2];
                sm_votes[base + 12] = d[3];
            }
        }
    }
    __syncthreads();

    // rr init = 1/O (reuse pose buffer)
    for (int idx = tid; idx < II * OO; idx += 256) sm_pose_rr[idx] = 1.0f / 16.0f;
    __syncthreads();
    float* rr = sm_pose_rr;

    // ---------------- Stage 3: EM routing ----------------
    const int o_t = tid >> 4;     // output capsule handled by this thread (M-step)
    const int p_t = tid & 15;     // pose element handled by this thread

    for (int t = 0; t < 3; ++t) {
        const float inv_temp = 1.0f + (float)t;   // 1 + (3-1)*t/(3-1)

        // ---- M-step ----
        float sum_rp = 0.0f, sum_rv = 0.0f;
        for (int i = 0; i < II; ++i) {
            float rp = rr[i * 16 + o_t] * sm_a[i];
            sum_rp += rp;
            sum_rv += rp * sm_votes[i * VSTR + o_t * 16 + p_t];
        }
        const float rinv = 1.0f / sum_rp;
        const float mean = sum_rv * rinv;
        float sum_v2 = 0.0f;
        for (int i = 0; i < II; ++i) {
            float rp = rr[i * 16 + o_t] * sm_a[i];
            float dv = sm_votes[i * VSTR + o_t * 16 + p_t] - mean;
            sum_v2 += rp * dv * dv;
        }
        const float var = sum_v2 * rinv;
        sm_mean[tid]   = mean;
        sm_inv2v[tid]  = 1.0f / (2.0f * var + EPS_F);
        sm_logstd[tid] = logf(sqrtf(var) + EPS_F);
        if (p_t == 0) sm_rsum[o_t] = sum_rp;
        __syncthreads();

        if (tid < 16) {   // per-o cost reduce + activation
            float ls = 0.0f;
            for (int p = 0; p < 16; ++p) ls += sm_logstd[tid * 16 + p];
            float costsum = sm_rsum[tid] * (16.0f * beta_v[tid] + ls);
            float x = inv_temp * (beta_a[tid] - costsum);
            float oact = 1.0f / (1.0f + expf(-x));
            sm_oact[tid]   = oact;
            sm_logact[tid] = logf(oact + EPS_F) - ls;  // hoists per-(i,o) logf out of E-step
        }
        __syncthreads();

        // ---- E-step ----
        if (t < 2) {
            if (tid < II) {
                float zz[16];
                float zmax = -1e30f;
                for (int o = 0; o < 16; ++o) {
                    float s = 0.0f;
                    const int bvv = tid * VSTR + o * 16;
                    const int bm  = o * 16;
                    for (int p = 0; p < 16; ++p) {
                        float dv = sm_votes[bvv + p] - sm_mean[bm + p];
                        s += dv * dv * sm_inv2v[bm + p];   // pure FMA chain now
                    }
                    float v = sm_logact[o] - s;
                    zz[o] = v;
                    zmax = fmaxf(zmax, v);
                }
                float esum = 0.0f;
                for (int o = 0; o < 16; ++o) { zz[o] = expf(zz[o] - zmax); esum += zz[o]; }
                float inv = 1.0f / esum;
                for (int o = 0; o < 16; ++o) rr[tid * 16 + o] = zz[o] * inv;
            }
            __syncthreads();
        }
    }

    // ---------------- Outputs ----------------
    out[n * 256 + tid] = sm_mean[tid];                                        // pose
    if (tid < 16) out[(BB * OHW * OHW) * 256 + n * 16 + tid] = sm_oact[tid];  // activation
}

extern "C" void kernel_launch(void* const* d_in, const int* in_sizes, int n_in,
                              void* d_out, int out_size, void* d_ws, size_t ws_size,
                              hipStream_t stream) {
    (void)in_sizes; (void)n_in; (void)d_ws; (void)ws_size; (void)out_size;
    const float* pose_in = (const float*)d_in[0];
    const float* act_in  = (const float*)d_in[1];
    const float* w       = (const float*)d_in[2];
    const float* beta_v  = (const float*)d_in[3];
    const float* beta_a  = (const float*)d_in[4];
    float* out = (float*)d_out;

    const int n_pos = BB * OHW * OHW;  // 1152 workgroups, one per output pixel
    capsule_em_fused<<<n_pos, 256, 0, stream>>>(pose_in, act_in, w, beta_v, beta_a, out);
}